// Line_81071802679625
// MI455X (gfx1250) — compile-verified
//
#include <hip/hip_runtime.h>
#include <math.h>

// ---------------------------------------------------------------------------
// Skip-gram negative-sampling loss on MI455X (gfx1250, wave32).
//
// Memory-bound: ~58.7 MB of gathered 512B embedding rows @ 23.3 TB/s -> ~2.5us.
// Dot products are batched onto V_WMMA_F32_16X16X4_F32 (exact f32 math):
// per 16-element tile, D = A(16xK) * B(Kx16); diag(D) = 16 pairwise dots.
// ---------------------------------------------------------------------------

typedef __attribute__((ext_vector_type(2))) float v2f;
typedef __attribute__((ext_vector_type(8))) float v8f;

#define EMBED_DIM 128
#define BATCH     16384
#define NUM_NEG   5
#define TILE      16
#define WAVES_PER_BLOCK 4
#define NUM_TILES (BATCH / TILE)   // 1024 waves total

__device__ __forceinline__ float log_sigmoid(float x) {
    // log(sigmoid(x)) = min(x,0) - log1p(exp(-|x|))   (numerically stable)
    return fminf(x, 0.0f) - log1pf(expf(-fabsf(x)));
}

__device__ __forceinline__ v8f wmma_f32(v2f a, v2f b, v8f c) {
    // 8 args: (neg_a, A, neg_b, B, c_mod, C, reuse_a, reuse_b)
    return __builtin_amdgcn_wmma_f32_16x16x4_f32(false, a, false, b,
                                                 (short)0, c, false, false);
}

__global__ void __launch_bounds__(WAVES_PER_BLOCK * 32)
nsloss_kernel(const float* __restrict__ emb,
              const int*   __restrict__ nodeindex,
              const int*   __restrict__ v_i,
              const int*   __restrict__ v_j,
              const int*   __restrict__ negs,     // [NUM_NEG, BATCH]
              float*       __restrict__ partials) // [NUM_TILES]
{
    const int lane = threadIdx.x & 31;            // wave32
    const int wave = threadIdx.x >> 5;
    const int tile = blockIdx.x * WAVES_PER_BLOCK + wave;
    const int b0   = tile * TILE;
    const int m    = lane & 15;                   // M-row (A) == N-col (B)
    const int half = lane >> 4;                   // K half-select within chunk
    const int b    = b0 + m;

    // Row pointers: honor the nodeindex indirection (identity in setup, but
    // correct for any permutation).  emb rows are 512B-contiguous gathers.
    const float* pa  = emb + (size_t)nodeindex[v_i[b]]             * EMBED_DIM;
    const float* pb  = emb + (size_t)nodeindex[v_j[b]]             * EMBED_DIM;
    const float* pn0 = emb + (size_t)nodeindex[negs[0 * BATCH + b]] * EMBED_DIM;
    const float* pn1 = emb + (size_t)nodeindex[negs[1 * BATCH + b]] * EMBED_DIM;
    const float* pn2 = emb + (size_t)nodeindex[negs[2 * BATCH + b]] * EMBED_DIM;
    const float* pn3 = emb + (size_t)nodeindex[negs[3 * BATCH + b]] * EMBED_DIM;
    const float* pn4 = emb + (size_t)nodeindex[negs[4 * BATCH + b]] * EMBED_DIM;

    v8f accP = {}, acc0 = {}, acc1 = {}, acc2 = {}, acc3 = {}, acc4 = {};

    // A 16x4 f32 fragment: lane m, VGPR0/1 = K = k0+2*half, k0+2*half+1.
    // B 4x16 f32 fragment: identical per-lane addressing (N = lane&15).
    #pragma unroll 4
    for (int k0 = 0; k0 < EMBED_DIM; k0 += 4) {
        const int off = k0 + 2 * half;
        v2f a  = *(const v2f*)(pa  + off);
        v2f bj = *(const v2f*)(pb  + off);
        v2f n0 = *(const v2f*)(pn0 + off);
        v2f n1 = *(const v2f*)(pn1 + off);
        v2f n2 = *(const v2f*)(pn2 + off);
        v2f n3 = *(const v2f*)(pn3 + off);
        v2f n4 = *(const v2f*)(pn4 + off);
        accP = wmma_f32(a, bj, accP);
        acc0 = wmma_f32(a, n0, acc0);
        acc1 = wmma_f32(a, n1, acc1);
        acc2 = wmma_f32(a, n2, acc2);
        acc3 = wmma_f32(a, n3, acc3);
        acc4 = wmma_f32(a, n4, acc4);
    }

    // Diagonal extraction (C/D layout, ISA 7.12.2):
    //  lanes 0-7  hold D[m][m]     at c[lane]      (M = lane)
    //  lanes 24-31 hold D[m][m]    at c[lane-24]   (M = lane-16)
    float loss = 0.0f;
    int d = -1;
    if (lane < 8)        d = lane;
    else if (lane >= 24) d = lane - 24;
    if (d >= 0) {
        loss  = log_sigmoid( accP[d]);
        loss += log_sigmoid(-acc0[d]);
        loss += log_sigmoid(-acc1[d]);
        loss += log_sigmoid(-acc2[d]);
        loss += log_sigmoid(-acc3[d]);
        loss += log_sigmoid(-acc4[d]);
    }

    // Wave-level butterfly reduction (inactive lanes contribute 0).
    #pragma unroll
    for (int o = 16; o > 0; o >>= 1)
        loss += __shfl_xor(loss, o, 32);

    if (lane == 0)
        partials[tile] = loss;   // fixed tile slot -> deterministic output
}

__global__ void reduce_kernel(const float* __restrict__ partials,
                              float* __restrict__ out)
{
    __shared__ double sdata[256];
    double s = 0.0;
    for (int i = threadIdx.x; i < NUM_TILES; i += 256)
        s += (double)partials[i];
    sdata[threadIdx.x] = s;
    __syncthreads();
    for (int stride = 128; stride > 0; stride >>= 1) {
        if ((int)threadIdx.x < stride)
            sdata[threadIdx.x] += sdata[threadIdx.x + stride];
        __syncthreads();
    }
    if (threadIdx.x == 0)
        out[0] = (float)(-sdata[0] / (double)BATCH);
}

extern "C" void kernel_launch(void* const* d_in, const int* in_sizes, int n_in,
                              void* d_out, int out_size, void* d_ws, size_t ws_size,
                              hipStream_t stream) {
    (void)in_sizes; (void)n_in; (void)out_size; (void)ws_size;
    const float* emb       = (const float*)d_in[0];
    const int*   nodeindex = (const int*)  d_in[1];
    const int*   v_i       = (const int*)  d_in[2];
    const int*   v_j       = (const int*)  d_in[3];
    const int*   negs      = (const int*)  d_in[4];
    float*       partials  = (float*)d_ws;         // 1024 floats of scratch

    nsloss_kernel<<<dim3(NUM_TILES / WAVES_PER_BLOCK),
                    dim3(WAVES_PER_BLOCK * 32), 0, stream>>>(
        emb, nodeindex, v_i, v_j, negs, partials);
    reduce_kernel<<<dim3(1), dim3(256), 0, stream>>>(partials, (float*)d_out);
}